// Convpass_56693568307653
// MI455X (gfx1250) — compile-verified
//
#include <hip/hip_runtime.h>
#include <hip/hip_bf16.h>

// Convpass adapter, 3-kernel pipeline for MI455X (gfx1250, wave32):
//   K1: x_down = quick_gelu(x @ Wd + bd)           [fp32 WMMA 16x16x4, K=768]
//   K2: a      = quick_gelu(conv3x3(x_down) + bc)  [VALU, Wc staged in LDS]
//   K3: out    = a @ Wu + bu                       [fp32 WMMA 16x16x4, K=8]
// Intermediates (2 MB each) live in d_ws and stay resident in the 192 MB L2.
// B fragments are pre-swizzled into LDS in exact WMMA lane layout so the hot
// loops contain only unconditional ds_load_b64/b128 + global_load_b64 + wmma.

typedef __attribute__((ext_vector_type(2))) float v2f;
typedef __attribute__((ext_vector_type(4))) float v4f;
typedef __attribute__((ext_vector_type(8))) float v8f;

#define BATCH 64
#define NTOK  1024
#define CDIM  768
#define ADIM  8
#define HGRID 32
#define KSTEPS (CDIM / 4)     // 192 k-steps for the down GEMM
#define NTILES (CDIM / 16)    // 48 column tiles for the up GEMM

__device__ __forceinline__ float qgelu(float v) {
    // v * sigmoid(1.702 v) = v * rcp(1 + exp(-1.702 v))
    // v_rcp_f32 instead of the IEEE divide sequence; plenty accurate here.
    return v * __builtin_amdgcn_rcpf(1.0f + __expf(-1.702f * v));
}

// ---------------------------------------------------------------------------
// K1: down projection + quick_gelu.
// One wave computes one 16x16 output tile (16 rows of x_down; cols 0..7 real,
// 8..15 zero-padded). Grid: 64 images * 64 row-tiles = 4096 waves.
// B fragments pre-swizzled: sB[step][lane][2], zero pad baked in.
// ---------------------------------------------------------------------------
__global__ void __launch_bounds__(256)
down_gelu_kernel(const float* __restrict__ x,
                 const float* __restrict__ Wd,
                 const float* __restrict__ bd,
                 float* __restrict__ g_ws) {
    __shared__ float sB[KSTEPS * 32 * 2];   // 48 KB, WMMA-lane layout
    __shared__ float sBd[ADIM];

    const int tid = threadIdx.x;
    for (int i = tid; i < KSTEPS * 32; i += 256) {
        const int step = i >> 5;
        const int L    = i & 31;
        const int h    = L >> 4;
        const int l    = L & 15;
        const int k    = step * 4 + 2 * h;      // B(k,n): k = step*4 + 2*hf + r
        v2f v;
        v.x = (l < ADIM) ? Wd[(k + 0) * ADIM + l] : 0.0f;
        v.y = (l < ADIM) ? Wd[(k + 1) * ADIM + l] : 0.0f;
        *reinterpret_cast<v2f*>(&sB[i * 2]) = v;
    }
    if (tid < ADIM) sBd[tid] = bd[tid];
    __syncthreads();

    const int wave = blockIdx.x * 8 + (tid >> 5);
    const int lane = tid & 31;
    const int hf   = lane >> 4;
    const int l16  = lane & 15;

    const int img = wave >> 6;          // / 64
    const int m0  = (wave & 63) * 16;   // row-tile base within image

    // A fragment: A(m,k), m = l16, k = k0 + 2*hf + r -> v2f at k0 + 2*hf.
    const float* xrow = x + ((size_t)img * NTOK + m0 + l16) * CDIM + 2 * hf;
    const float* bbase = sB + lane * 2;  // + step*64

    v8f acc0 = {}, acc1 = {};
    #pragma unroll 2
    for (int k0 = 0; k0 < CDIM; k0 += 8) {
        v2f a0 = *reinterpret_cast<const v2f*>(xrow + k0);
        v2f b0 = *reinterpret_cast<const v2f*>(bbase + (k0 >> 2) * 64);
        v2f a1 = *reinterpret_cast<const v2f*>(xrow + k0 + 4);
        v2f b1 = *reinterpret_cast<const v2f*>(bbase + (k0 >> 2) * 64 + 64);
        acc0 = __builtin_amdgcn_wmma_f32_16x16x4_f32(
            false, a0, false, b0, (short)0, acc0, false, false);
        acc1 = __builtin_amdgcn_wmma_f32_16x16x4_f32(
            false, a1, false, b1, (short)0, acc1, false, false);
    }

    // D(m,n): n = l16, m = v + 8*hf. Keep only n < 8.
    if (l16 < ADIM) {
        const float bias = sBd[l16];
        float* gbase = g_ws + ((size_t)img * NTOK + m0 + 8 * hf) * ADIM + l16;
        #pragma unroll
        for (int v = 0; v < 8; ++v)
            gbase[(size_t)v * ADIM] = qgelu(acc0[v] + acc1[v] + bias);
    }
}

// ---------------------------------------------------------------------------
// K2: 3x3 SAME conv (dim 8 -> 8) + quick_gelu. One thread = one pixel,
// all 8 output channels. 65536 pixels -> 256 blocks of 256.
// ---------------------------------------------------------------------------
__global__ void __launch_bounds__(256)
conv_gelu_kernel(const float* __restrict__ g_ws,
                 const float* __restrict__ Wc,
                 const float* __restrict__ bc,
                 float* __restrict__ a_ws) {
    __shared__ float sWc[9 * ADIM * ADIM];   // 2304 B
    __shared__ float sBc[ADIM];

    const int tid = threadIdx.x;
    for (int i = tid; i < 9 * ADIM * ADIM; i += 256) sWc[i] = Wc[i];
    if (tid < ADIM) sBc[tid] = bc[tid];
    __syncthreads();

    const int p   = blockIdx.x * 256 + tid;      // global pixel id
    const int img = p >> 10;
    const int pi  = p & (NTOK - 1);
    const int y   = pi >> 5;
    const int xx  = pi & (HGRID - 1);

    float acc[ADIM];
    #pragma unroll
    for (int co = 0; co < ADIM; ++co) acc[co] = 0.0f;

    const float* gimg = g_ws + (size_t)img * NTOK * ADIM;
    #pragma unroll
    for (int ky = 0; ky < 3; ++ky) {
        const int iy = y + ky - 1;
        if (iy < 0 || iy >= HGRID) continue;
        #pragma unroll
        for (int kx = 0; kx < 3; ++kx) {
            const int ix = xx + kx - 1;
            if (ix < 0 || ix >= HGRID) continue;
            const v4f* gp = reinterpret_cast<const v4f*>(
                gimg + (size_t)(iy * HGRID + ix) * ADIM);
            const v4f gv0 = gp[0];
            const v4f gv1 = gp[1];
            const float* wp = sWc + (ky * 3 + kx) * ADIM * ADIM;
            #pragma unroll
            for (int ci = 0; ci < 4; ++ci) {
                #pragma unroll
                for (int co = 0; co < ADIM; ++co) {
                    acc[co] = fmaf(gv0[ci], wp[ci * ADIM + co], acc[co]);
                    acc[co] = fmaf(gv1[ci], wp[(ci + 4) * ADIM + co], acc[co]);
                }
            }
        }
    }

    v4f o0, o1;
    #pragma unroll
    for (int co = 0; co < 4; ++co) o0[co] = qgelu(acc[co] + sBc[co]);
    #pragma unroll
    for (int co = 0; co < 4; ++co) o1[co] = qgelu(acc[co + 4] + sBc[co + 4]);
    v4f* dst = reinterpret_cast<v4f*>(a_ws + (size_t)p * ADIM);
    dst[0] = o0;
    dst[1] = o1;
}

// ---------------------------------------------------------------------------
// K3: up projection. One wave owns one 16-row tile of one image and loops
// over 48 column tiles; two chained WMMAs give K=8.
// B fragments pre-swizzled: sBf[ntile][lane][4] -> one ds_load_b128 per tile.
// Grid: 64 images * 64 row-tiles = 4096 waves.
// ---------------------------------------------------------------------------
__global__ void __launch_bounds__(256)
up_kernel(const float* __restrict__ a_ws,
          const float* __restrict__ Wu,
          const float* __restrict__ bu,
          float* __restrict__ out) {
    __shared__ float sBf[NTILES * 32 * 4];   // 24 KB, WMMA-lane layout
    __shared__ float sBu[CDIM];              // 3 KB

    const int tid = threadIdx.x;
    for (int i = tid; i < NTILES * 32; i += 256) {
        const int nt = i >> 5;
        const int L  = i & 31;
        const int h  = L >> 4;
        const int l  = L & 15;
        const int n  = nt * 16 + l;
        const int kb = 2 * h;                 // B(k,n): k = 4*s + 2*hf + r
        v4f v;
        v.x = Wu[(kb + 0) * CDIM + n];
        v.y = Wu[(kb + 1) * CDIM + n];
        v.z = Wu[(kb + 4) * CDIM + n];
        v.w = Wu[(kb + 5) * CDIM + n];
        *reinterpret_cast<v4f*>(&sBf[i * 4]) = v;
    }
    for (int i = tid; i < CDIM; i += 256) sBu[i] = bu[i];
    __syncthreads();

    const int wave = blockIdx.x * 8 + (tid >> 5);
    const int lane = tid & 31;
    const int hf   = lane >> 4;
    const int l16  = lane & 15;

    const int img = wave >> 6;
    const int m0  = (wave & 63) * 16;

    // A fragments (reused across all n-tiles): A(m,k), m = l16,
    // k = 4*s + 2*hf + r  -> v2f load at k = 4*s + 2*hf (8B aligned).
    const float* arow = a_ws + ((size_t)img * NTOK + m0 + l16) * ADIM + 2 * hf;
    const v2f af0 = *reinterpret_cast<const v2f*>(arow + 0);
    const v2f af1 = *reinterpret_cast<const v2f*>(arow + 4);

    const float* bfbase = sBf + lane * 4;    // + nt*128
    float* orow = out + ((size_t)img * NTOK + m0 + 8 * hf) * CDIM + l16;

    for (int nt = 0; nt < NTILES; ++nt) {
        const v4f bq = *reinterpret_cast<const v4f*>(bfbase + nt * 128);
        v2f bf0, bf1;
        bf0.x = bq.x; bf0.y = bq.y;
        bf1.x = bq.z; bf1.y = bq.w;

        v8f acc = {};
        acc = __builtin_amdgcn_wmma_f32_16x16x4_f32(
            false, af0, false, bf0, (short)0, acc, false, false);
        acc = __builtin_amdgcn_wmma_f32_16x16x4_f32(
            false, af1, false, bf1, (short)0, acc, false, false);

        const float bias = sBu[nt * 16 + l16];
        #pragma unroll
        for (int v = 0; v < 8; ++v)
            orow[(size_t)v * CDIM + nt * 16] = acc[v] + bias;
    }
}

// ---------------------------------------------------------------------------
extern "C" void kernel_launch(void* const* d_in, const int* in_sizes, int n_in,
                              void* d_out, int out_size, void* d_ws, size_t ws_size,
                              hipStream_t stream) {
    (void)in_sizes; (void)n_in; (void)out_size; (void)ws_size;
    const float* x  = (const float*)d_in[0];
    const float* Wd = (const float*)d_in[1];
    const float* bd = (const float*)d_in[2];
    const float* Wc = (const float*)d_in[3];
    const float* bc = (const float*)d_in[4];
    const float* Wu = (const float*)d_in[5];
    const float* bu = (const float*)d_in[6];
    float* out  = (float*)d_out;
    float* g_ws = (float*)d_ws;                                  // 2 MB
    float* a_ws = g_ws + (size_t)BATCH * NTOK * ADIM;            // 2 MB

    // 4096 waves each for K1/K3 -> 512 blocks of 8 waves (wave32).
    down_gelu_kernel<<<512, 256, 0, stream>>>(x, Wd, bd, g_ws);
    conv_gelu_kernel<<<256, 256, 0, stream>>>(g_ws, Wc, bc, a_ws);
    up_kernel<<<512, 256, 0, stream>>>(a_ws, Wu, bu, out);
}